// MagicAttn_56478819943055
// MI455X (gfx1250) — compile-verified
//
#include <hip/hip_runtime.h>
#include <math.h>

#define N_NODES  50000
#define N_EDGES  800000
#define N_HEADS  4
#define HEAD_DIM 32
#define FDIM     128            // N_HEADS * HEAD_DIM
#define ALPHA    0.2f

typedef float v2f __attribute__((ext_vector_type(2)));
typedef float v8f __attribute__((ext_vector_type(8)));

// Order-preserving float <-> uint mapping (monotone for all finite values, -inf ok)
__device__ __forceinline__ unsigned f2ord(float f) {
    unsigned u = __float_as_uint(f);
    return (u & 0x80000000u) ? ~u : (u | 0x80000000u);
}
__device__ __forceinline__ float ord2f(unsigned u) {
    return (u & 0x80000000u) ? __uint_as_float(u & 0x7fffffffu)
                             : __uint_as_float(~u);
}

// ---------------- init: zero output, init segment max (-inf) / sum (0) -------
__global__ void gat_init_kernel(float* __restrict__ out,
                                unsigned* __restrict__ segmax,
                                float* __restrict__ segsum) {
    int i = blockIdx.x * blockDim.x + threadIdx.x;
    if (i < N_NODES * FDIM) out[i] = 0.0f;
    if (i < N_NODES * N_HEADS) {
        segmax[i] = 0x007FFFFFu;   // f2ord(-inf)
        segsum[i] = 0.0f;
    }
}

// ---------------- pass 1: logits via V_WMMA_F32_16X16X4_F32 ------------------
// One wave handles a 16-edge tile. A = 16x4 f32 slab of edge_emb, B = 4x16
// one-hot-by-head matrix carrying attn_w in column h (h = chunk/8). Chaining
// 32 K=4 WMMAs accumulates the full 128-dim per-head dot products exactly in
// f32. Leaky-ReLU applied, then atomic segment-max keyed by dst.
__global__ void __launch_bounds__(256)
gat_logits_wmma_kernel(const float* __restrict__ edge_emb,
                       const float* __restrict__ attn_w,   // 128 floats: w[h*32+d]
                       const int*   __restrict__ dst,
                       float*       __restrict__ logits,   // [E, 4]
                       unsigned*    __restrict__ segmax) { // [N, 4] ordered-uint
    __shared__ float sw[FDIM];
    int t = threadIdx.x;
    if (t < FDIM) sw[t] = attn_w[t];
    __syncthreads();

    int wave = (blockIdx.x * blockDim.x + t) >> 5;
    int lane = t & 31;
    int e0   = wave * 16;
    if (e0 >= N_EDGES) return;            // uniform per wave (E % 16 == 0)

    int hi  = lane >> 4;                  // half-wave: K pairs {0,1} vs {2,3}
    int row = lane & 15;                  // A: M row   | B: N column
    const float* arow = edge_emb + (size_t)(e0 + row) * FDIM + 2 * hi;

    v8f acc = {0.f, 0.f, 0.f, 0.f, 0.f, 0.f, 0.f, 0.f};
#pragma unroll
    for (int c = 0; c < 32; ++c) {
        v2f a  = *(const v2f*)(arow + 4 * c);          // K = 4c+2*hi, +1
        v2f bw = *(const v2f*)(&sw[4 * c + 2 * hi]);   // matching attn_w pair
        int h  = c >> 3;                               // active output column
        v2f b;
        b.x = (row == h) ? bw.x : 0.0f;                // one-hot column h
        b.y = (row == h) ? bw.y : 0.0f;
        acc = __builtin_amdgcn_wmma_f32_16x16x4_f32(
            /*neg_a=*/false, a, /*neg_b=*/false, b,
            /*c_mod=*/(short)0, acc, /*reuse_a=*/false, /*reuse_b=*/false);
    }

    // D layout: acc[v] = D[v + 8*hi][row]. Columns 0..3 carry the head logits.
    if (row < N_HEADS) {
        int h = row;
#pragma unroll
        for (int v = 0; v < 8; ++v) {
            int   e = e0 + v + 8 * hi;
            float x = acc[v];
            x = (x >= 0.0f) ? x : ALPHA * x;           // leaky_relu
            logits[e * N_HEADS + h] = x;
            atomicMax(&segmax[(size_t)dst[e] * N_HEADS + h], f2ord(x));
        }
    }
}

// ---------------- pass 2: exp(logit - segmax[dst]) + segment-sum -------------
__global__ void gat_expsum_kernel(const int* __restrict__ dst,
                                  const unsigned* __restrict__ segmax,
                                  float* __restrict__ logits_ex,  // in: logits, out: ex
                                  float* __restrict__ segsum) {
    int t = blockIdx.x * blockDim.x + threadIdx.x;
    if (t >= N_EDGES * N_HEADS) return;
    int e = t >> 2, h = t & 3;
    int d = dst[e];
    float m  = ord2f(segmax[d * N_HEADS + h]);
    float ex = __expf(logits_ex[t] - m);
    logits_ex[t] = ex;
    atomicAdd(&segsum[d * N_HEADS + h], ex);
}

// ---------------- pass 3: normalize, gather src, scatter-add by dst ----------
// One wave per edge: 32 lanes x float4 = the full 128-float row. Lane's head
// h = lane/8. 512B global_load_b128 per matrix, 4 f32 atomics per lane to L2
// (out array fits in the 192MB L2).
__global__ void __launch_bounds__(256)
gat_scatter_kernel(const float* __restrict__ edge_emb,
                   const float* __restrict__ node_emb,
                   const int*   __restrict__ src,
                   const int*   __restrict__ dst,
                   const float* __restrict__ ex,       // [E, 4]
                   const float* __restrict__ segsum,   // [N, 4]
                   float*       __restrict__ out) {    // [N, 128]
    int t = blockIdx.x * blockDim.x + threadIdx.x;
    int e = t >> 5;
    if (e >= N_EDGES) return;                          // uniform per wave
    int lane = t & 31;
    int d = dst[e], s = src[e];
    int h = lane >> 3;                                 // (lane*4)/32
    float attn = ex[e * N_HEADS + h] / segsum[(size_t)d * N_HEADS + h];

    float4 ev = *((const float4*)(edge_emb + (size_t)e * FDIM) + lane);
    float4 nv = *((const float4*)(node_emb + (size_t)s * FDIM) + lane);
    float* op = out + (size_t)d * FDIM + lane * 4;
    atomicAdd(op + 0, attn * ev.x * nv.x);
    atomicAdd(op + 1, attn * ev.y * nv.y);
    atomicAdd(op + 2, attn * ev.z * nv.z);
    atomicAdd(op + 3, attn * ev.w * nv.w);
}

extern "C" void kernel_launch(void* const* d_in, const int* in_sizes, int n_in,
                              void* d_out, int out_size, void* d_ws, size_t ws_size,
                              hipStream_t stream) {
    const float* edge_emb = (const float*)d_in[0];   // [E, 128]
    const float* node_emb = (const float*)d_in[1];   // [N, 128]
    const float* attn_w   = (const float*)d_in[2];   // [4, 32, 1] -> 128 floats
    const int*   src      = (const int*)d_in[3];     // [E]
    const int*   dst      = (const int*)d_in[4];     // [E]
    float*       out      = (float*)d_out;           // [N, 128]

    // workspace: logits/ex [E*4] f32 | segmax [N*4] u32 | segsum [N*4] f32
    float*    logits = (float*)d_ws;
    unsigned* segmax = (unsigned*)(logits + (size_t)N_EDGES * N_HEADS);
    float*    segsum = (float*)(segmax + (size_t)N_NODES * N_HEADS);

    gat_init_kernel<<<(N_NODES * FDIM + 255) / 256, 256, 0, stream>>>(out, segmax, segsum);

    // 50000 16-edge tiles, 1 wave each, 8 waves per block
    gat_logits_wmma_kernel<<<(N_EDGES / 16 * 32 + 255) / 256, 256, 0, stream>>>(
        edge_emb, attn_w, dst, logits, segmax);

    gat_expsum_kernel<<<(N_EDGES * N_HEADS + 255) / 256, 256, 0, stream>>>(
        dst, segmax, logits, segsum);

    gat_scatter_kernel<<<((size_t)N_EDGES * 32 + 255) / 256, 256, 0, stream>>>(
        edge_emb, node_emb, src, dst, logits, segsum, out);
}